// IMUMambaEncoder_41618233098305
// MI455X (gfx1250) — compile-verified
//
#include <hip/hip_runtime.h>
#include <hip/hip_bf16.h>

#define BATCH 8
#define SEQ   4096
#define DM    128
#define DI    256
#define DST   16
#define CH    64   // chunks
#define CT    64   // chunk length (CH*CT == SEQ)

typedef __attribute__((ext_vector_type(2))) float v2f;
typedef __attribute__((ext_vector_type(8))) float v8f;

__device__ __forceinline__ v8f wmma_f32_4(v2f a, v2f b, v8f c) {
  // V_WMMA_F32_16X16X4_F32: D = A(16x4) * B(4x16) + C(16x16), fp32 exact
  return __builtin_amdgcn_wmma_f32_16x16x4_f32(
      /*neg_a=*/false, a, /*neg_b=*/false, b,
      /*c_mod=*/(short)0, c, /*reuse_a=*/false, /*reuse_b=*/false);
}

__device__ __forceinline__ float sigmoidf_fast(float x) {
  return 1.0f / (1.0f + __expf(-x));
}

// ---------------------------------------------------------------------------
// Kernel 1: front-end MLP + LayerNorm + in-projection, one 16-token tile/block
//   h1 = relu(x*w1^T+b1); h2 = h1*w2^T+b2 (WMMA); hn = LN(h2);
//   xz = hn*in_w^T (WMMA) -> xx, z ; resid = h2
// ---------------------------------------------------------------------------
__global__ __launch_bounds__(256) void k_frontend(
    const float* __restrict__ x, const float* __restrict__ w1,
    const float* __restrict__ b1, const float* __restrict__ w2,
    const float* __restrict__ b2, const float* __restrict__ ln_g,
    const float* __restrict__ ln_b, const float* __restrict__ in_w,
    float* __restrict__ resid, float* __restrict__ xx, float* __restrict__ zz) {
  __shared__ float h1[16][129];   // +1 pad: avoid 16-way LDS bank conflicts
  __shared__ float hn[16][129];
  const long tok0 = (long)blockIdx.x * 16;
  const int  tid  = threadIdx.x;

  // ---- h1 = relu(x * w1^T + b1), K=6 (tiny, VALU) ----
  for (int i = tid; i < 16 * DM; i += 256) {
    int m = i >> 7, o = i & 127;
    const float* xp = x + (tok0 + m) * 6;
    float acc = b1[o];
#pragma unroll
    for (int k = 0; k < 6; ++k) acc = fmaf(xp[k], w1[o * 6 + k], acc);
    h1[m][o] = fmaxf(acc, 0.0f);
  }
  __syncthreads();

  const int wave = tid >> 5, lane = tid & 31;
  const int lr = lane & 15, half = lane >> 4;

  // ---- h2 = h1 * w2^T + b2, each wave owns 16 output cols ----
  {
    const int n0 = wave * 16;
    v8f c;
    const float bias = b2[n0 + lr];
#pragma unroll
    for (int v = 0; v < 8; ++v) c[v] = bias;
    const float* wrow = w2 + (long)(n0 + lr) * DM;
    for (int k0 = 0; k0 < DM; k0 += 4) {
      v2f a, bf;
      a[0]  = h1[lr][k0 + 2 * half + 0];
      a[1]  = h1[lr][k0 + 2 * half + 1];
      bf[0] = wrow[k0 + 2 * half + 0];
      bf[1] = wrow[k0 + 2 * half + 1];
      c = wmma_f32_4(a, bf, c);
    }
#pragma unroll
    for (int v = 0; v < 8; ++v) {
      int m = v + 8 * half;
      hn[m][n0 + lr] = c[v];
      resid[(tok0 + m) * DM + n0 + lr] = c[v];
    }
  }
  __syncthreads();

  // ---- LayerNorm each of the 16 token rows ----
  if (tid < 16) {
    float mu = 0.0f;
    for (int k = 0; k < DM; ++k) mu += hn[tid][k];
    mu *= (1.0f / DM);
    float var = 0.0f;
    for (int k = 0; k < DM; ++k) { float d = hn[tid][k] - mu; var = fmaf(d, d, var); }
    var *= (1.0f / DM);
    float inv = rsqrtf(var + 1e-5f);
    for (int k = 0; k < DM; ++k)
      hn[tid][k] = (hn[tid][k] - mu) * inv * ln_g[k] + ln_b[k];
  }
  __syncthreads();

  // ---- xz = hn * in_w^T (16 x 512), 4 tiles per wave ----
  for (int j = 0; j < 4; ++j) {
    const int n0 = (wave * 4 + j) * 16;
    v8f c;
#pragma unroll
    for (int v = 0; v < 8; ++v) c[v] = 0.0f;
    const float* wrow = in_w + (long)(n0 + lr) * DM;
    for (int k0 = 0; k0 < DM; k0 += 4) {
      v2f a, bf;
      a[0]  = hn[lr][k0 + 2 * half + 0];
      a[1]  = hn[lr][k0 + 2 * half + 1];
      bf[0] = wrow[k0 + 2 * half + 0];
      bf[1] = wrow[k0 + 2 * half + 1];
      c = wmma_f32_4(a, bf, c);
    }
    const int e = n0 + lr;
#pragma unroll
    for (int v = 0; v < 8; ++v) {
      int m = v + 8 * half;
      if (e < DI) xx[(tok0 + m) * DI + e]        = c[v];
      else        zz[(tok0 + m) * DI + (e - DI)] = c[v];
    }
  }
}

// ---------------------------------------------------------------------------
// Kernel 2: causal depthwise conv (K=4) + SiLU -> u
// ---------------------------------------------------------------------------
__global__ __launch_bounds__(256) void k_conv_silu(
    const float* __restrict__ xx, const float* __restrict__ conv_w,
    const float* __restrict__ conv_b, float* __restrict__ u) {
  long idx = (long)blockIdx.x * 256 + threadIdx.x;   // (b*L+l)*DI + d
  int  d   = (int)(idx & (DI - 1));
  long bl  = idx >> 8;
  int  l   = (int)(bl & (SEQ - 1));
  float acc = conv_b[d];
#pragma unroll
  for (int j = 0; j < 4; ++j) {
    int ll = l - 3 + j;
    if (ll >= 0) acc = fmaf(conv_w[d * 4 + j], xx[(bl - 3 + j) * DI + d], acc);
  }
  u[idx] = acc * sigmoidf_fast(acc);
}

// ---------------------------------------------------------------------------
// Kernel 3: x-projection (256->40) + dt head (8->256, softplus, clip)
// One token per 128-thread block.
// ---------------------------------------------------------------------------
__global__ __launch_bounds__(128) void k_xproj(
    const float* __restrict__ u, const float* __restrict__ xproj_w,
    const float* __restrict__ dt_w, const float* __restrict__ dt_b,
    float* __restrict__ dtg, float* __restrict__ Bm, float* __restrict__ Cm) {
  __shared__ float us[DI];
  __shared__ float xd[40];
  const long tok = blockIdx.x;
  const int  tid = threadIdx.x;
  us[tid]       = u[tok * DI + tid];
  us[tid + 128] = u[tok * DI + tid + 128];
  __syncthreads();
  if (tid < 40) {
    const float* wrow = xproj_w + tid * DI;
    float acc = 0.0f;
    for (int k = 0; k < DI; ++k) acc = fmaf(us[k], wrow[k], acc);
    xd[tid] = acc;
    if (tid >= 8 && tid < 24)      Bm[tok * DST + tid - 8]  = acc;
    else if (tid >= 24)            Cm[tok * DST + tid - 24] = acc;
  }
  __syncthreads();
#pragma unroll
  for (int r = 0; r < 2; ++r) {
    int d = tid + r * 128;
    float acc = dt_b[d];
#pragma unroll
    for (int k = 0; k < 8; ++k) acc = fmaf(xd[k], dt_w[d * 8 + k], acc);
    float sp = (acc > 20.0f) ? acc : __logf(1.0f + __expf(acc));
    dtg[tok * DI + d] = fminf(sp, 100.0f);
  }
}

// ---------------------------------------------------------------------------
// Kernel 4a: chunked scan pass 1 — local chunk state (h0=0) and sum(dt).
// Block = one (b, chunk); thread = channel d; B-vectors staged in LDS.
// ---------------------------------------------------------------------------
__global__ __launch_bounds__(256) void k_scan_pass1(
    const float* __restrict__ dtg, const float* __restrict__ ug,
    const float* __restrict__ Bm, const float* __restrict__ A_log,
    float* __restrict__ S, float* __restrict__ sumdt) {
  __shared__ float Bs[CT][DST];
  const int bc = blockIdx.x;           // b*CH + c
  const int b  = bc / CH, c = bc % CH;
  const int d  = threadIdx.x;
  const long l0 = (long)b * SEQ + (long)c * CT;
  for (int i = threadIdx.x; i < CT * DST; i += 256)
    Bs[i >> 4][i & 15] = Bm[(l0 + (i >> 4)) * DST + (i & 15)];
  __syncthreads();
  float A[DST], h[DST];
#pragma unroll
  for (int n = 0; n < DST; ++n) { A[n] = -__expf(A_log[d * DST + n]); h[n] = 0.0f; }
  float sdt = 0.0f;
  for (int t = 0; t < CT; ++t) {
    __builtin_prefetch(&dtg[(l0 + t + 8) * DI + d], 0, 0);
    __builtin_prefetch(&ug [(l0 + t + 8) * DI + d], 0, 0);
    float dt = dtg[(l0 + t) * DI + d];
    float uu = ug [(l0 + t) * DI + d];
    sdt += dt;
    float du = dt * uu;
#pragma unroll
    for (int n = 0; n < DST; ++n)
      h[n] = fmaf(__expf(dt * A[n]), h[n], du * Bs[t][n]);
  }
  const long o = (long)bc * DI + d;
  sumdt[o] = sdt;
#pragma unroll
  for (int n = 0; n < DST; ++n) S[o * DST + n] = h[n];
}

// ---------------------------------------------------------------------------
// Kernel 4b: sequential prefix across the 64 chunks (per (b,d,n) state).
// H_c = exp(A*sumdt_c)*H_{c-1} + S_c ; records chunk start states.
// ---------------------------------------------------------------------------
__global__ __launch_bounds__(256) void k_scan_prefix(
    const float* __restrict__ S, const float* __restrict__ sumdt,
    const float* __restrict__ A_log, float* __restrict__ Hstart) {
  int idx = blockIdx.x * 256 + threadIdx.x;   // b*DI*DST total = 32768
  int n = idx & 15;
  int d = (idx >> 4) & 255;
  int b = idx >> 12;
  float A = -__expf(A_log[d * DST + n]);
  float H = 0.0f;
  for (int c = 0; c < CH; ++c) {
    long o = (long)(b * CH + c) * DI + d;
    Hstart[o * DST + n] = H;
    H = fmaf(__expf(A * sumdt[o]), H, S[o * DST + n]);
  }
}

// ---------------------------------------------------------------------------
// Kernel 4c: chunked scan pass 2 — replay chunk with correct start state,
// emit y_t = <h,C_t> + u*D_skip, gate with silu(z).
// ---------------------------------------------------------------------------
__global__ __launch_bounds__(256) void k_scan_pass2(
    const float* __restrict__ dtg, const float* __restrict__ ug,
    const float* __restrict__ Bm, const float* __restrict__ Cm,
    const float* __restrict__ zg, const float* __restrict__ A_log,
    const float* __restrict__ D_skip, const float* __restrict__ Hstart,
    float* __restrict__ yg) {
  __shared__ float Bs[CT][DST];
  __shared__ float Cs[CT][DST];
  const int bc = blockIdx.x;
  const int b  = bc / CH, c = bc % CH;
  const int d  = threadIdx.x;
  const long l0 = (long)b * SEQ + (long)c * CT;
  for (int i = threadIdx.x; i < CT * DST; i += 256) {
    Bs[i >> 4][i & 15] = Bm[(l0 + (i >> 4)) * DST + (i & 15)];
    Cs[i >> 4][i & 15] = Cm[(l0 + (i >> 4)) * DST + (i & 15)];
  }
  __syncthreads();
  const long o = (long)bc * DI + d;
  float A[DST], h[DST];
#pragma unroll
  for (int n = 0; n < DST; ++n) {
    A[n] = -__expf(A_log[d * DST + n]);
    h[n] = Hstart[o * DST + n];
  }
  const float Dk = D_skip[d];
  for (int t = 0; t < CT; ++t) {
    __builtin_prefetch(&dtg[(l0 + t + 8) * DI + d], 0, 0);
    __builtin_prefetch(&ug [(l0 + t + 8) * DI + d], 0, 0);
    float dt = dtg[(l0 + t) * DI + d];
    float uu = ug [(l0 + t) * DI + d];
    float du = dt * uu;
    float y = 0.0f;
#pragma unroll
    for (int n = 0; n < DST; ++n) {
      h[n] = fmaf(__expf(dt * A[n]), h[n], du * Bs[t][n]);
      y = fmaf(h[n], Cs[t][n], y);
    }
    y = fmaf(uu, Dk, y);
    float zv = zg[(l0 + t) * DI + d];
    y *= zv * sigmoidf_fast(zv);
    yg[(l0 + t) * DI + d] = y;
  }
}

// ---------------------------------------------------------------------------
// Kernel 5: out-projection y(16xDI) * out_w^T(DIxDM) + resid  (WMMA, K=256)
// ---------------------------------------------------------------------------
__global__ __launch_bounds__(256) void k_outproj(
    const float* __restrict__ yg, const float* __restrict__ out_w,
    const float* __restrict__ resid, float* __restrict__ out) {
  __shared__ float ys[16][257];   // pad: avoid bank conflicts on A-frag reads
  const long tok0 = (long)blockIdx.x * 16;
  const int  tid  = threadIdx.x;
  for (int i = tid; i < 16 * DI; i += 256) {
    int m = i >> 8, k = i & 255;
    ys[m][k] = yg[(tok0 + m) * DI + k];
  }
  __syncthreads();
  const int wave = tid >> 5, lane = tid & 31;
  const int lr = lane & 15, half = lane >> 4;
  const int n0 = wave * 16;
  v8f c;
#pragma unroll
  for (int v = 0; v < 8; ++v) c[v] = 0.0f;
  const float* wrow = out_w + (long)(n0 + lr) * DI;
  for (int k0 = 0; k0 < DI; k0 += 4) {
    v2f a, bf;
    a[0]  = ys[lr][k0 + 2 * half + 0];
    a[1]  = ys[lr][k0 + 2 * half + 1];
    bf[0] = wrow[k0 + 2 * half + 0];
    bf[1] = wrow[k0 + 2 * half + 1];
    c = wmma_f32_4(a, bf, c);
  }
#pragma unroll
  for (int v = 0; v < 8; ++v) {
    int m = v + 8 * half;
    long idx = (tok0 + m) * DM + n0 + lr;
    out[idx] = c[v] + resid[idx];
  }
}

// Kernel 6: deterministic fill of the second tuple output (host-RNG diag,
// unreproducible on device) — write zeros.
__global__ void k_diag_zero(float* __restrict__ diag) {
  diag[blockIdx.x * 256 + threadIdx.x] = 0.0f;
}

// ---------------------------------------------------------------------------
extern "C" void kernel_launch(void* const* d_in, const int* in_sizes, int n_in,
                              void* d_out, int out_size, void* d_ws, size_t ws_size,
                              hipStream_t stream) {
  const float* x       = (const float*)d_in[0];
  const float* w1      = (const float*)d_in[1];
  const float* b1      = (const float*)d_in[2];
  const float* w2      = (const float*)d_in[3];
  const float* b2      = (const float*)d_in[4];
  const float* ln_g    = (const float*)d_in[5];
  const float* ln_b    = (const float*)d_in[6];
  const float* in_w    = (const float*)d_in[7];
  const float* conv_w  = (const float*)d_in[8];
  const float* conv_b  = (const float*)d_in[9];
  const float* xproj_w = (const float*)d_in[10];
  const float* dt_w    = (const float*)d_in[11];
  const float* dt_b    = (const float*)d_in[12];
  const float* A_log   = (const float*)d_in[13];
  const float* D_skip  = (const float*)d_in[14];
  const float* out_w   = (const float*)d_in[15];

  float* out  = (float*)d_out;                      // (B,L,DM) flat
  float* diag = out + (long)BATCH * SEQ * DM;       // (DM,DM) flat

  // workspace partition (floats)
  float* ws     = (float*)d_ws;
  const long NT = (long)BATCH * SEQ;                // 32768 tokens
  float* resid  = ws;                 ws += NT * DM;        // 16.8 MB
  float* xx     = ws;                 ws += NT * DI;        // 33.6 MB
  float* zz     = ws;                 ws += NT * DI;
  float* u      = ws;                 ws += NT * DI;
  float* dtg    = ws;                 ws += NT * DI;
  float* Bm     = ws;                 ws += NT * DST;
  float* Cm     = ws;                 ws += NT * DST;
  float* yg     = ws;                 ws += NT * DI;
  float* S      = ws;                 ws += (long)BATCH * CH * DI * DST;
  float* sumdt  = ws;                 ws += (long)BATCH * CH * DI;
  float* Hstart = ws;                 /* last */            // total ~206 MB

  const int tiles = (int)(NT / 16);                 // 2048

  k_frontend<<<tiles, 256, 0, stream>>>(x, w1, b1, w2, b2, ln_g, ln_b, in_w,
                                        resid, xx, zz);
  k_conv_silu<<<(int)(NT * DI / 256), 256, 0, stream>>>(xx, conv_w, conv_b, u);
  k_xproj<<<(int)NT, 128, 0, stream>>>(u, xproj_w, dt_w, dt_b, dtg, Bm, Cm);
  k_scan_pass1<<<BATCH * CH, 256, 0, stream>>>(dtg, u, Bm, A_log, S, sumdt);
  k_scan_prefix<<<(BATCH * DI * DST) / 256, 256, 0, stream>>>(S, sumdt, A_log, Hstart);
  k_scan_pass2<<<BATCH * CH, 256, 0, stream>>>(dtg, u, Bm, Cm, zz, A_log, D_skip,
                                               Hstart, yg);
  k_outproj<<<tiles, 256, 0, stream>>>(yg, out_w, resid, out);
  k_diag_zero<<<(DM * DM) / 256, 256, 0, stream>>>(diag);
}